// TaskTaskGATkLayer_90881507983881
// MI455X (gfx1250) — compile-verified
//
#include <hip/hip_runtime.h>

#define HID 16
#define LRELU_ATT 0.2f
#define LRELU_OUT 0.01f
#define LN_EPS 1e-5f

typedef float v2f __attribute__((ext_vector_type(2)));
typedef float v8f __attribute__((ext_vector_type(8)));

__device__ __forceinline__ float lrelu(float v, float s) { return v > 0.0f ? v : v * s; }

// ---------------------------------------------------------------------------
// Projection: for a 16-row node tile, compute
//   xl = x @ Wl^T + bl ; xr = x @ Wr^T + br ; xres = x @ Wres^T + bias
// using V_WMMA_F32_16X16X4_F32 (full f32, matches reference precision).
// One wave per tile; three accumulators share the A operand.
// ---------------------------------------------------------------------------
__global__ void proj_wmma_kernel(const float* __restrict__ x, int Cin, int N,
                                 const float* __restrict__ Wl, const float* __restrict__ bl,
                                 const float* __restrict__ Wr, const float* __restrict__ brv,
                                 const float* __restrict__ Wres, const float* __restrict__ bias,
                                 float* __restrict__ xl, float* __restrict__ xr,
                                 float* __restrict__ xres)
{
    const int lane  = threadIdx.x & 31;
    const int wave  = threadIdx.x >> 5;
    const int tile  = blockIdx.x * (blockDim.x >> 5) + wave;
    const int ntile = (N + 15) >> 4;
    if (tile >= ntile) return;               // wave-uniform: EXEC stays all-1s for WMMA

    const int row0  = tile << 4;
    const int mrow  = lane & 15;             // A-matrix row M
    const int khalf = lane >> 4;             // K sub-pair select: {0,1} vs {2,3}
    const int col   = lane & 15;             // B/D column N (output channel)

    int arow = row0 + mrow;
    if (arow >= N) arow = N - 1;             // clamped (N is a multiple of 16 here anyway)
    const float* __restrict__ xrow = x + (size_t)arow * (size_t)Cin;

    v8f accL = {}, accR = {}, accS = {};

    for (int k = 0; k < Cin; k += 4) {
        const int ko = k + 2 * khalf;        // identical per-lane offset for A and B=W^T
        v2f a, bL, bR, bS;
        a.x  = xrow[ko];                a.y  = xrow[ko + 1];
        bL.x = Wl  [col * Cin + ko];    bL.y = Wl  [col * Cin + ko + 1];
        bR.x = Wr  [col * Cin + ko];    bR.y = Wr  [col * Cin + ko + 1];
        bS.x = Wres[col * Cin + ko];    bS.y = Wres[col * Cin + ko + 1];
        accL = __builtin_amdgcn_wmma_f32_16x16x4_f32(false, a, false, bL, (short)0, accL, false, false);
        accR = __builtin_amdgcn_wmma_f32_16x16x4_f32(false, a, false, bR, (short)0, accR, false, false);
        accS = __builtin_amdgcn_wmma_f32_16x16x4_f32(false, a, false, bS, (short)0, accS, false, false);
    }

    const float cbl = bl[col], cbr = brv[col], cbb = bias[col];
    #pragma unroll
    for (int v = 0; v < 8; ++v) {
        const int r = row0 + v + 8 * khalf;  // D layout: vgpr v, lane half -> row
        if (r < N) {
            const size_t o = (size_t)r * HID + col;
            xl[o]   = accL[v] + cbl;
            xr[o]   = accR[v] + cbr;
            xres[o] = accS[v] + cbb;
        }
    }
}

// ---------------------------------------------------------------------------
// Per-phase node-state init: accum=0, denom=0, smax=-inf (bit pattern)
// ---------------------------------------------------------------------------
__global__ void init_node_kernel(int* __restrict__ smax_bits, float* __restrict__ denom,
                                 float* __restrict__ accum, int N)
{
    int i = blockIdx.x * blockDim.x + threadIdx.x;
    if (i < N * HID) accum[i] = 0.0f;
    if (i < N) { smax_bits[i] = (int)0xFF800000u; denom[i] = 0.0f; }
}

// attention logit for one edge: sum_c leaky(xl[s][c]+xr[d][c], 0.2) * att[c]
__device__ __forceinline__ float edge_score(const float* __restrict__ xl,
                                            const float* __restrict__ xr,
                                            const float* __restrict__ att,
                                            int s, int d)
{
    const float4* pl = (const float4*)(xl + (size_t)s * HID);
    const float4* pr = (const float4*)(xr + (size_t)d * HID);
    const float4* pw = (const float4*)att;
    float sc = 0.0f;
    #pragma unroll
    for (int q = 0; q < 4; ++q) {
        float4 a = pl[q], b = pr[q], w = pw[q];
        sc += lrelu(a.x + b.x, LRELU_ATT) * w.x;
        sc += lrelu(a.y + b.y, LRELU_ATT) * w.y;
        sc += lrelu(a.z + b.z, LRELU_ATT) * w.z;
        sc += lrelu(a.w + b.w, LRELU_ATT) * w.w;
    }
    return sc;
}

// Pass 1: segment max over dst (order-preserving int/uint atomic trick -> HW atomics)
__global__ void edge_max_kernel(const int* __restrict__ src, const int* __restrict__ dst,
                                const float* __restrict__ xl, const float* __restrict__ xr,
                                const float* __restrict__ att, int* __restrict__ smax_bits, int E)
{
    int e = blockIdx.x * blockDim.x + threadIdx.x;
    if (e >= E) return;
    const int s = src[e], d = dst[e];
    const float sc = edge_score(xl, xr, att, s, d);
    const int bits = __float_as_int(sc);
    if (bits >= 0) atomicMax(&smax_bits[d], bits);
    else           atomicMin((unsigned int*)&smax_bits[d], (unsigned int)bits);
}

// Pass 2: denom[dst] += exp(s - smax[dst])
__global__ void edge_expsum_kernel(const int* __restrict__ src, const int* __restrict__ dst,
                                   const float* __restrict__ xl, const float* __restrict__ xr,
                                   const float* __restrict__ att, const float* __restrict__ smax,
                                   float* __restrict__ denom, int E)
{
    int e = blockIdx.x * blockDim.x + threadIdx.x;
    if (e >= E) return;
    const int s = src[e], d = dst[e];
    const float sc = edge_score(xl, xr, att, s, d);
    const float a = __expf(sc - smax[d]);
    unsafeAtomicAdd(&denom[d], a);           // -> GLOBAL_ATOMIC_ADD_F32
}

// Pass 3: accum[dst] += alpha * xl[src]
__global__ void edge_scatter_kernel(const int* __restrict__ src, const int* __restrict__ dst,
                                    const float* __restrict__ xl, const float* __restrict__ xr,
                                    const float* __restrict__ att, const float* __restrict__ smax,
                                    const float* __restrict__ denom, float* __restrict__ accum, int E)
{
    int e = blockIdx.x * blockDim.x + threadIdx.x;
    if (e >= E) return;
    const int s = src[e], d = dst[e];
    const float sc = edge_score(xl, xr, att, s, d);
    const float alpha = __expf(sc - smax[d]) / (denom[d] + 1e-16f);
    const float4* pl = (const float4*)(xl + (size_t)s * HID);
    float* base = accum + (size_t)d * HID;
    #pragma unroll
    for (int q = 0; q < 4; ++q) {
        float4 a = pl[q];
        unsafeAtomicAdd(base + 4 * q + 0, alpha * a.x);
        unsafeAtomicAdd(base + 4 * q + 1, alpha * a.y);
        unsafeAtomicAdd(base + 4 * q + 2, alpha * a.z);
        unsafeAtomicAdd(base + 4 * q + 3, alpha * a.w);
    }
}

// out = leaky( LN(accum + xres) , 0.01 )   (xres already carries +bias)
__global__ void finish_kernel(const float* __restrict__ accum, const float* __restrict__ xres,
                              const float* __restrict__ lnw, const float* __restrict__ lnb,
                              float* __restrict__ outx, int N)
{
    int n = blockIdx.x * blockDim.x + threadIdx.x;
    if (n >= N) return;
    float y[HID];
    float mu = 0.0f;
    #pragma unroll
    for (int c = 0; c < HID; ++c) {
        y[c] = accum[(size_t)n * HID + c] + xres[(size_t)n * HID + c];
        mu += y[c];
    }
    mu *= (1.0f / HID);
    float var = 0.0f;
    #pragma unroll
    for (int c = 0; c < HID; ++c) { float t = y[c] - mu; var += t * t; }
    var *= (1.0f / HID);
    const float inv = rsqrtf(var + LN_EPS);
    #pragma unroll
    for (int c = 0; c < HID; ++c) {
        float v = (y[c] - mu) * inv * lnw[c] + lnb[c];
        outx[(size_t)n * HID + c] = lrelu(v, LRELU_OUT);
    }
}

// final concat: [dep(16) | dpt(16) | tasks(64)]
__global__ void concat_kernel(const float* __restrict__ dep, const float* __restrict__ dpt,
                              const float* __restrict__ tasks, float* __restrict__ out, int N)
{
    int i = blockIdx.x * blockDim.x + threadIdx.x;
    if (i >= N * 96) return;
    const int n = i / 96, c = i % 96;
    float v;
    if (c < 16)      v = dep[(size_t)n * 16 + c];
    else if (c < 32) v = dpt[(size_t)n * 16 + (c - 16)];
    else             v = tasks[(size_t)n * 64 + (c - 32)];
    out[i] = v;
}

// ---------------------------------------------------------------------------
// d_in layout (flat dict order):
//  0: task_embedding [N,64]
//  1..7 : dep_params[0] {Wl,bl,Wr,br,att,Wres,bias}   8..14: dep_params[1]
// 15..21: dpt_params[0]                              22..28: dpt_params[1]
// 29/30: ln_dep_w/b   31/32: ln_dpt_w/b   33: edge_index [2,E] int32
// ---------------------------------------------------------------------------
extern "C" void kernel_launch(void* const* d_in, const int* in_sizes, int n_in,
                              void* d_out, int out_size, void* d_ws, size_t ws_size,
                              hipStream_t stream)
{
    const float* tasks = (const float*)d_in[0];
    const int N = in_sizes[0] / 64;
    const int* edges = (const int*)d_in[33];
    const int E = in_sizes[33] / 2;
    const int* srcA = edges;
    const int* dstA = edges + E;

    float* ws = (float*)d_ws;
    float* xl    = ws; ws += (size_t)N * HID;
    float* xr    = ws; ws += (size_t)N * HID;
    float* xres  = ws; ws += (size_t)N * HID;
    float* accum = ws; ws += (size_t)N * HID;
    float* curd  = ws; ws += (size_t)N * HID;
    float* curp  = ws; ws += (size_t)N * HID;
    float* denom = ws; ws += (size_t)N;
    int*   smaxb = (int*)ws;
    float* cur[2] = { curd, curp };

    const float* lnw[2] = { (const float*)d_in[29], (const float*)d_in[31] };
    const float* lnb[2] = { (const float*)d_in[30], (const float*)d_in[32] };

    const int nodeBlocks = (N + 255) / 256;
    const int nhBlocks   = (N * HID + 255) / 256;
    const int edgeBlocks = (E + 255) / 256;
    const int wavesPerBlk = 8;                         // 256 threads = 8 wave32
    const int ntiles     = (N + 15) / 16;
    const int projBlocks = (ntiles + wavesPerBlk - 1) / wavesPerBlk;

    for (int layer = 0; layer < 2; ++layer) {
        const int Cin = (layer == 0) ? 64 : HID;
        for (int br = 0; br < 2; ++br) {
            const int pbase = 1 + br * 14 + layer * 7;
            const float* Wl   = (const float*)d_in[pbase + 0];
            const float* bl   = (const float*)d_in[pbase + 1];
            const float* Wr   = (const float*)d_in[pbase + 2];
            const float* brb  = (const float*)d_in[pbase + 3];
            const float* att  = (const float*)d_in[pbase + 4];
            const float* Wres = (const float*)d_in[pbase + 5];
            const float* bias = (const float*)d_in[pbase + 6];
            const float* xin  = (layer == 0) ? tasks : cur[br];
            // dpt branch runs GATv2 with flipped edges
            const int* se = (br == 0) ? srcA : dstA;
            const int* de = (br == 0) ? dstA : srcA;

            proj_wmma_kernel<<<projBlocks, 32 * wavesPerBlk, 0, stream>>>(
                xin, Cin, N, Wl, bl, Wr, brb, Wres, bias, xl, xr, xres);
            init_node_kernel<<<nhBlocks, 256, 0, stream>>>(smaxb, denom, accum, N);
            edge_max_kernel<<<edgeBlocks, 256, 0, stream>>>(se, de, xl, xr, att, smaxb, E);
            edge_expsum_kernel<<<edgeBlocks, 256, 0, stream>>>(se, de, xl, xr, att,
                (const float*)smaxb, denom, E);
            edge_scatter_kernel<<<edgeBlocks, 256, 0, stream>>>(se, de, xl, xr, att,
                (const float*)smaxb, denom, accum, E);
            finish_kernel<<<nodeBlocks, 256, 0, stream>>>(accum, xres, lnw[br], lnb[br], cur[br], N);
        }
    }

    const int total = N * 96;
    concat_kernel<<<(total + 255) / 256, 256, 0, stream>>>(curd, curp, tasks, (float*)d_out, N);
}